// SpikingLinear_su_61890478735915
// MI455X (gfx1250) — compile-verified
//
#include <hip/hip_runtime.h>

typedef __attribute__((ext_vector_type(2))) float v2f;
typedef __attribute__((ext_vector_type(8))) float v8f;

#define STEPS 128
#define BATCH 256
#define D1    1024
#define D2    1024
#define MROWS ((STEPS - 1) * BATCH)   // 32512
#define BM 128
#define BN 128
#define BK 32
#define LDSROW 36                     // padded row stride (dwords): conflict-free + 16B aligned

// ---------------------------------------------------------------------------
// Phase 1: lin[k] = input[k] @ W^T, written into the I-output at time plane k+1.
// Flattened: Out[(r + BATCH)*D2 + n] = sum_k In[r*D1 + k] * W[n*D1 + k]
// ---------------------------------------------------------------------------
__global__ __launch_bounds__(256) void snn_gemm_f32_wmma(
    const float* __restrict__ input,   // [STEPS*BATCH, D1] (uses first MROWS rows)
    const float* __restrict__ weight,  // [D2, D1]
    float* __restrict__ iout)          // base of I region: [STEPS, BATCH, D2]
{
    __shared__ float As[BM * LDSROW];
    __shared__ float Bs[BN * LDSROW];

    const int tid  = threadIdx.x;
    const int lane = tid & 31;
    const int wave = tid >> 5;
    const int wm   = wave & 3;         // 4 waves along M: 32 rows each
    const int wn   = wave >> 2;        // 2 waves along N: 64 cols each
    const int rowBase = blockIdx.y * BM;
    const int colBase = blockIdx.x * BN;

    const int l16   = lane & 15;
    const int khalf = (lane >> 4) << 1;   // 0 for lanes 0-15, 2 for lanes 16-31

    v8f acc[2][4];
    #pragma unroll
    for (int mi = 0; mi < 2; ++mi)
        #pragma unroll
        for (int ni = 0; ni < 4; ++ni)
            acc[mi][ni] = (v8f){};

    for (int kt = 0; kt < D1; kt += BK) {
        // Cooperative tile fill: 128x32 f32 each for A and B, float4 per thread x4.
        #pragma unroll
        for (int i = 0; i < 4; ++i) {
            int slot = tid + i * 256;
            int r    = slot >> 3;
            int c4   = (slot & 7) << 2;
            float4 av = *(const float4*)(input  + (size_t)(rowBase + r) * D1 + kt + c4);
            *(float4*)(&As[r * LDSROW + c4]) = av;
            float4 bv = *(const float4*)(weight + (size_t)(colBase + r) * D1 + kt + c4);
            *(float4*)(&Bs[r * LDSROW + c4]) = bv;
        }
        __syncthreads();

        #pragma unroll
        for (int kk = 0; kk < BK; kk += 4) {
            const int ko = kk + khalf;
            v2f a[2], b[4];
            #pragma unroll
            for (int mi = 0; mi < 2; ++mi) {
                int r = wm * 32 + mi * 16 + l16;
                a[mi] = *(const v2f*)(&As[r * LDSROW + ko]);
            }
            #pragma unroll
            for (int ni = 0; ni < 4; ++ni) {
                int c = wn * 64 + ni * 16 + l16;
                b[ni] = *(const v2f*)(&Bs[c * LDSROW + ko]);
            }
            #pragma unroll
            for (int mi = 0; mi < 2; ++mi)
                #pragma unroll
                for (int ni = 0; ni < 4; ++ni)
                    acc[mi][ni] = __builtin_amdgcn_wmma_f32_16x16x4_f32(
                        false, a[mi], false, b[ni],
                        (short)0, acc[mi][ni], false, false);
        }
        __syncthreads();
    }

    // Store C. VGPR j holds row j (lanes 0-15) / row j+8 (lanes 16-31), col = lane%16.
    // Shift by one time plane (k -> t = k+1) == +BATCH rows.
    float* cbase = iout + (size_t)BATCH * D2;
    #pragma unroll
    for (int mi = 0; mi < 2; ++mi) {
        const int rbase = rowBase + wm * 32 + mi * 16 + ((lane >> 4) << 3);
        #pragma unroll
        for (int ni = 0; ni < 4; ++ni) {
            const int col = colBase + wn * 64 + ni * 16 + l16;
            #pragma unroll
            for (int j = 0; j < 8; ++j)
                cbase[(size_t)(rbase + j) * D2 + col] = acc[mi][ni][j];
        }
    }
}

// ---------------------------------------------------------------------------
// Phase 2: sequential synaptic/membrane recurrence. One thread per (b, d2).
// Reads lin (staged in I region) and overwrites it in place with syn.
// ---------------------------------------------------------------------------
__global__ __launch_bounds__(256) void snn_scan(
    float* __restrict__ spk, float* __restrict__ V, float* __restrict__ I)
{
    const float ALPHA = 0.7788007830714049f;   // exp(-1/4)
    const float BETA  = 0.9512294245007140f;   // exp(-1/20)
    const size_t plane = (size_t)BATCH * D2;
    const size_t idx = (size_t)blockIdx.x * blockDim.x + threadIdx.x;

    // t = 0 planes are zero.
    spk[idx] = 0.0f;  V[idx] = 0.0f;  I[idx] = 0.0f;

    float syn = 0.0f, mem = 0.0f;
    for (int t = 1; t < STEPS; ++t) {
        const size_t off = (size_t)t * plane + idx;
        const float cur   = I[off];                       // staged lin[t-1]
        const float reset = (mem > 1.0f) ? 1.0f : 0.0f;   // reset from PREVIOUS mem
        syn = ALPHA * syn + cur;
        mem = (BETA * mem + syn) * (1.0f - reset);
        const float s = (mem > 1.0f) ? 1.0f : 0.0f;
        spk[off] = s;
        V[off]   = mem;
        I[off]   = syn;
    }
}

extern "C" void kernel_launch(void* const* d_in, const int* in_sizes, int n_in,
                              void* d_out, int out_size, void* d_ws, size_t ws_size,
                              hipStream_t stream) {
    const float* input  = (const float*)d_in[0];   // [STEPS, BATCH, D1]
    const float* weight = (const float*)d_in[1];   // [D2, D1]
    float* out = (float*)d_out;

    const size_t tplane = (size_t)STEPS * BATCH * D2;
    float* spk = out;                 // output[0]
    float* V   = out + tplane;        // output[1]
    float* I   = out + 2 * tplane;    // output[2] (also GEMM staging)

    snn_gemm_f32_wmma<<<dim3(D2 / BN, MROWS / BM), 256, 0, stream>>>(input, weight, I);
    snn_scan<<<dim3((BATCH * D2) / 256), 256, 0, stream>>>(spk, V, I);
}